// SwiGLUExperts_69733089017993
// MI455X (gfx1250) — compile-verified
//
#include <hip/hip_runtime.h>
#include <hip/hip_bf16.h>
#include <math.h>

// MoE SwiGLU experts for MI455X (gfx1250), fp32 via V_WMMA_F32_16X16X4_F32.
//   h   = silu(x @ W1e[:, :I]) * (x @ W1e[:, I:])   (kernel 1, fused epilogue)
//   out = h @ W2e                                    (kernel 2)
//
// - Per-expert token counts share gcd 128 -> BM=128 row tiles never straddle
//   an expert boundary; each block resolves its expert with an 8-entry scan.
// - B tiles stored TRANSPOSED in LDS ([n][k], even stride) so every WMMA
//   operand fragment is a single aligned ds_load_b64 (no v_mov reassembly).
// - Register-staged double buffering: next K-tile's global_load_b128s issue
//   before the WMMA chain, LDS stores after it; one barrier per K-tile.

typedef __attribute__((ext_vector_type(2))) float v2f;
typedef __attribute__((ext_vector_type(8))) float v8f;

#define NEXPERTS 8
#define HDIM 1024
#define IDIM 1024

#define BM 128
#define BK 32
#define LDA 34   // even padded stride: aligned b64 frag loads, odd/2 bank step
#define LDB 34

// ---- WMMA f32 16x16x4:  D = A(16x4) * B(4x16) + C ----
__device__ inline v8f wmma4(v2f a, v2f b, v8f c) {
    return __builtin_amdgcn_wmma_f32_16x16x4_f32(
        false, a, false, b, (short)0, c, false, false);
}

__device__ inline int find_expert(const int* __restrict__ counts, int row0) {
    int off = 0;
    int e = NEXPERTS - 1;
    #pragma unroll
    for (int i = 0; i < NEXPERTS; ++i) {
        int c = counts[i];
        if (row0 < off + c) { e = i; break; }
        off += c;
    }
    return e;
}

// Store one float4 as two aligned v2f (8B) LDS stores (dst is 8B aligned).
__device__ inline void st_lds4(float* dst, float4 v) {
    v2f lo, hi;
    lo.x = v.x; lo.y = v.y; hi.x = v.z; hi.y = v.w;
    *(v2f*)dst = lo;
    *(v2f*)(dst + 2) = hi;
}

// ---------------- Kernel 1: gate/up GEMM + SwiGLU -> h[T][I] ----------------
// Block tile: 128 rows x 64 gate cols + 64 up cols. 8 waves in 4x2; each wave
// owns a 32x32 patch of BOTH gate and up (8 WMMA per K=4 step, 6 b64 loads).
__global__ __launch_bounds__(256)
void moe_gemm1_swiglu(const float* __restrict__ x,       // [T][H]
                      const float* __restrict__ w1,      // [E][H][2I]
                      const int*   __restrict__ counts,  // [E]
                      float*       __restrict__ h)       // [T][I] workspace
{
    __shared__ float As[2][BM * LDA];       // [m][k]
    __shared__ float Bg[2][64 * LDB];       // transposed: [n][k]
    __shared__ float Bu[2][64 * LDB];       // transposed: [n][k]

    const int tid   = threadIdx.x;
    const int lane  = tid & 31;
    const int wave  = tid >> 5;
    const int waveM = wave >> 1;            // 0..3
    const int waveN = wave & 1;             // 0..1

    const int rowBase = blockIdx.y * BM;
    const int colBase = blockIdx.x * 64;    // gate col; up col = +IDIM

    const int e = find_expert(counts, rowBase);
    const float* w1e = w1 + (size_t)e * HDIM * (2 * IDIM);

    v8f accG[2][2], accU[2][2];
    #pragma unroll
    for (int mi = 0; mi < 2; ++mi)
        #pragma unroll
        for (int ni = 0; ni < 2; ++ni) { accG[mi][ni] = (v8f){}; accU[mi][ni] = (v8f){}; }

    // Staging coordinates (fixed per thread)
    const int ar[4]  = { (tid + 0) >> 3, (tid + 256) >> 3, (tid + 512) >> 3, (tid + 768) >> 3 };
    const int ac4    = tid & 7;
    const int br[2]  = { tid >> 4, (tid + 256) >> 4 };
    const int bc4    = tid & 15;

    float4 aR[4], gR[2], uR[2];

    // Prologue: fetch K-tile 0
    #pragma unroll
    for (int i = 0; i < 4; ++i)
        aR[i] = *(const float4*)(x + (size_t)(rowBase + ar[i]) * HDIM + ac4 * 4);
    #pragma unroll
    for (int i = 0; i < 2; ++i) {
        const float* w = w1e + (size_t)br[i] * (2 * IDIM) + colBase + bc4 * 4;
        gR[i] = *(const float4*)(w);
        uR[i] = *(const float4*)(w + IDIM);
    }
    #pragma unroll
    for (int i = 0; i < 4; ++i) st_lds4(&As[0][ar[i] * LDA + ac4 * 4], aR[i]);
    #pragma unroll
    for (int i = 0; i < 2; ++i) {
        int nb = bc4 * 4;
        Bg[0][(nb + 0) * LDB + br[i]] = gR[i].x;
        Bg[0][(nb + 1) * LDB + br[i]] = gR[i].y;
        Bg[0][(nb + 2) * LDB + br[i]] = gR[i].z;
        Bg[0][(nb + 3) * LDB + br[i]] = gR[i].w;
        Bu[0][(nb + 0) * LDB + br[i]] = uR[i].x;
        Bu[0][(nb + 1) * LDB + br[i]] = uR[i].y;
        Bu[0][(nb + 2) * LDB + br[i]] = uR[i].z;
        Bu[0][(nb + 3) * LDB + br[i]] = uR[i].w;
    }

    const int nTiles = HDIM / BK;
    for (int t = 0; t < nTiles; ++t) {
        __syncthreads();
        const int cur = t & 1;
        const bool hasNext = (t + 1) < nTiles;

        // Issue next tile's global loads first (latency hidden by WMMAs)
        if (hasNext) {
            const int k0 = (t + 1) * BK;
            #pragma unroll
            for (int i = 0; i < 4; ++i)
                aR[i] = *(const float4*)(x + (size_t)(rowBase + ar[i]) * HDIM + k0 + ac4 * 4);
            #pragma unroll
            for (int i = 0; i < 2; ++i) {
                const float* w = w1e + (size_t)(k0 + br[i]) * (2 * IDIM) + colBase + bc4 * 4;
                gR[i] = *(const float4*)(w);
                uR[i] = *(const float4*)(w + IDIM);
            }
        }

        // Compute on current buffer
        const float* aBase = &As[cur][(waveM * 32 + (lane & 15)) * LDA + ((lane >> 4) << 1)];
        const float* gBase = &Bg[cur][(waveN * 32 + (lane & 15)) * LDB + ((lane >> 4) << 1)];
        const float* uBase = &Bu[cur][(waveN * 32 + (lane & 15)) * LDB + ((lane >> 4) << 1)];
        #pragma unroll
        for (int kk = 0; kk < BK; kk += 4) {
            v2f a0 = *(const v2f*)(aBase + kk);
            v2f a1 = *(const v2f*)(aBase + 16 * LDA + kk);
            v2f g0 = *(const v2f*)(gBase + kk);
            v2f g1 = *(const v2f*)(gBase + 16 * LDB + kk);
            v2f u0 = *(const v2f*)(uBase + kk);
            v2f u1 = *(const v2f*)(uBase + 16 * LDB + kk);

            accG[0][0] = wmma4(a0, g0, accG[0][0]);
            accG[0][1] = wmma4(a0, g1, accG[0][1]);
            accG[1][0] = wmma4(a1, g0, accG[1][0]);
            accG[1][1] = wmma4(a1, g1, accG[1][1]);
            accU[0][0] = wmma4(a0, u0, accU[0][0]);
            accU[0][1] = wmma4(a0, u1, accU[0][1]);
            accU[1][0] = wmma4(a1, u0, accU[1][0]);
            accU[1][1] = wmma4(a1, u1, accU[1][1]);
        }

        // Park next tile into the other LDS buffer
        if (hasNext) {
            const int nxt = cur ^ 1;
            #pragma unroll
            for (int i = 0; i < 4; ++i) st_lds4(&As[nxt][ar[i] * LDA + ac4 * 4], aR[i]);
            #pragma unroll
            for (int i = 0; i < 2; ++i) {
                int nb = bc4 * 4;
                Bg[nxt][(nb + 0) * LDB + br[i]] = gR[i].x;
                Bg[nxt][(nb + 1) * LDB + br[i]] = gR[i].y;
                Bg[nxt][(nb + 2) * LDB + br[i]] = gR[i].z;
                Bg[nxt][(nb + 3) * LDB + br[i]] = gR[i].w;
                Bu[nxt][(nb + 0) * LDB + br[i]] = uR[i].x;
                Bu[nxt][(nb + 1) * LDB + br[i]] = uR[i].y;
                Bu[nxt][(nb + 2) * LDB + br[i]] = uR[i].z;
                Bu[nxt][(nb + 3) * LDB + br[i]] = uR[i].w;
            }
        }
    }

    // Epilogue: h = silu(gate) * up (C/D layout: VGPR r -> M = r + 8*(lane/16))
    #pragma unroll
    for (int mi = 0; mi < 2; ++mi) {
        #pragma unroll
        for (int ni = 0; ni < 2; ++ni) {
            int n     = colBase + waveN * 32 + ni * 16 + (lane & 15);
            int rbase = rowBase + waveM * 32 + mi * 16 + ((lane >> 4) << 3);
            v8f g = accG[mi][ni];
            v8f u = accU[mi][ni];
            #pragma unroll
            for (int r = 0; r < 8; ++r) {
                float gv = g[r];
                float hv = (gv / (1.0f + __expf(-gv))) * u[r];
                h[(size_t)(rbase + r) * IDIM + n] = hv;
            }
        }
    }
}

// ---------------- Kernel 2: out[T][H] = h @ W2e ----------------
// Block tile: 128 x 128. 8 waves in 4x2; each wave owns 32 rows x 64 cols
// (2 A frags x 4 B frags -> 8 WMMA per K=4 step, 6 b64 loads).
__global__ __launch_bounds__(256)
void moe_gemm2(const float* __restrict__ h,       // [T][I]
               const float* __restrict__ w2,      // [E][I][H]
               const int*   __restrict__ counts,  // [E]
               float*       __restrict__ out)     // [T][H]
{
    __shared__ float As[2][BM * LDA];       // [m][k]
    __shared__ float Bs[2][128 * LDB];      // transposed: [n][k]

    const int tid   = threadIdx.x;
    const int lane  = tid & 31;
    const int wave  = tid >> 5;
    const int waveM = wave >> 1;            // 0..3 -> 32-row strip
    const int waveN = wave & 1;             // 0..1 -> 64-col strip

    const int rowBase = blockIdx.y * BM;
    const int colBase = blockIdx.x * 128;

    const int e = find_expert(counts, rowBase);
    const float* w2e = w2 + (size_t)e * IDIM * HDIM;

    v8f acc[2][4];
    #pragma unroll
    for (int mi = 0; mi < 2; ++mi)
        #pragma unroll
        for (int ni = 0; ni < 4; ++ni) acc[mi][ni] = (v8f){};

    const int ar[4] = { (tid + 0) >> 3, (tid + 256) >> 3, (tid + 512) >> 3, (tid + 768) >> 3 };
    const int ac4   = tid & 7;
    const int br[4] = { (tid + 0) >> 5, (tid + 256) >> 5, (tid + 512) >> 5, (tid + 768) >> 5 };
    const int bc4   = tid & 31;

    float4 aR[4], bR[4];

    // Prologue: K-tile 0
    #pragma unroll
    for (int i = 0; i < 4; ++i)
        aR[i] = *(const float4*)(h + (size_t)(rowBase + ar[i]) * IDIM + ac4 * 4);
    #pragma unroll
    for (int i = 0; i < 4; ++i)
        bR[i] = *(const float4*)(w2e + (size_t)br[i] * HDIM + colBase + bc4 * 4);
    #pragma unroll
    for (int i = 0; i < 4; ++i) st_lds4(&As[0][ar[i] * LDA + ac4 * 4], aR[i]);
    #pragma unroll
    for (int i = 0; i < 4; ++i) {
        int nb = bc4 * 4;
        Bs[0][(nb + 0) * LDB + br[i]] = bR[i].x;
        Bs[0][(nb + 1) * LDB + br[i]] = bR[i].y;
        Bs[0][(nb + 2) * LDB + br[i]] = bR[i].z;
        Bs[0][(nb + 3) * LDB + br[i]] = bR[i].w;
    }

    const int nTiles = IDIM / BK;
    for (int t = 0; t < nTiles; ++t) {
        __syncthreads();
        const int cur = t & 1;
        const bool hasNext = (t + 1) < nTiles;

        if (hasNext) {
            const int k0 = (t + 1) * BK;
            #pragma unroll
            for (int i = 0; i < 4; ++i)
                aR[i] = *(const float4*)(h + (size_t)(rowBase + ar[i]) * IDIM + k0 + ac4 * 4);
            #pragma unroll
            for (int i = 0; i < 4; ++i)
                bR[i] = *(const float4*)(w2e + (size_t)(k0 + br[i]) * HDIM + colBase + bc4 * 4);
        }

        const float* aBase = &As[cur][(waveM * 32 + (lane & 15)) * LDA + ((lane >> 4) << 1)];
        const float* bBase = &Bs[cur][(waveN * 64 + (lane & 15)) * LDB + ((lane >> 4) << 1)];
        #pragma unroll
        for (int kk = 0; kk < BK; kk += 4) {
            v2f a0 = *(const v2f*)(aBase + kk);
            v2f a1 = *(const v2f*)(aBase + 16 * LDA + kk);
            v2f b0 = *(const v2f*)(bBase + kk);
            v2f b1 = *(const v2f*)(bBase + 16 * LDB + kk);
            v2f b2 = *(const v2f*)(bBase + 32 * LDB + kk);
            v2f b3 = *(const v2f*)(bBase + 48 * LDB + kk);

            acc[0][0] = wmma4(a0, b0, acc[0][0]);
            acc[0][1] = wmma4(a0, b1, acc[0][1]);
            acc[0][2] = wmma4(a0, b2, acc[0][2]);
            acc[0][3] = wmma4(a0, b3, acc[0][3]);
            acc[1][0] = wmma4(a1, b0, acc[1][0]);
            acc[1][1] = wmma4(a1, b1, acc[1][1]);
            acc[1][2] = wmma4(a1, b2, acc[1][2]);
            acc[1][3] = wmma4(a1, b3, acc[1][3]);
        }

        if (hasNext) {
            const int nxt = cur ^ 1;
            #pragma unroll
            for (int i = 0; i < 4; ++i) st_lds4(&As[nxt][ar[i] * LDA + ac4 * 4], aR[i]);
            #pragma unroll
            for (int i = 0; i < 4; ++i) {
                int nb = bc4 * 4;
                Bs[nxt][(nb + 0) * LDB + br[i]] = bR[i].x;
                Bs[nxt][(nb + 1) * LDB + br[i]] = bR[i].y;
                Bs[nxt][(nb + 2) * LDB + br[i]] = bR[i].z;
                Bs[nxt][(nb + 3) * LDB + br[i]] = bR[i].w;
            }
        }
    }

    #pragma unroll
    for (int mi = 0; mi < 2; ++mi) {
        #pragma unroll
        for (int ni = 0; ni < 4; ++ni) {
            int n     = colBase + waveN * 64 + ni * 16 + (lane & 15);
            int rbase = rowBase + waveM * 32 + mi * 16 + ((lane >> 4) << 3);
            v8f c = acc[mi][ni];
            #pragma unroll
            for (int r = 0; r < 8; ++r)
                __builtin_nontemporal_store(c[r], &out[(size_t)(rbase + r) * HDIM + n]);
        }
    }
}

extern "C" void kernel_launch(void* const* d_in, const int* in_sizes, int n_in,
                              void* d_out, int out_size, void* d_ws, size_t ws_size,
                              hipStream_t stream) {
    const float* x      = (const float*)d_in[0];   // [T][H]
    const float* w1     = (const float*)d_in[1];   // [E][H][2I]
    const float* w2     = (const float*)d_in[2];   // [E][I][H]
    const int*   counts = (const int*)d_in[3];     // [E]
    float*       out    = (float*)d_out;           // [T][H]
    float*       hbuf   = (float*)d_ws;            // [T][I] scratch (32 MB)

    const int T = in_sizes[0] / HDIM;              // 8192

    dim3 block(256);
    dim3 g1(IDIM / 64, T / BM);                    // 16 x 64
    moe_gemm1_swiglu<<<g1, block, 0, stream>>>(x, w1, counts, hbuf);

    dim3 g2(HDIM / 128, T / BM);                   // 8 x 64
    moe_gemm2<<<g2, block, 0, stream>>>(hbuf, w2, counts, out);
}